// QuantumLayer_65481071400567
// MI455X (gfx1250) — compile-verified
//
#include <hip/hip_runtime.h>
#include <hip/hip_bf16.h>

// ============================================================================
// 14-qubit batched statevector simulator for MI455X (gfx1250).
//
// Kernel 1 (qsim_kernel): one workgroup per batch sample. The full 2^14
//   complex64 statevector (128 KB) lives in dynamic LDS (320 KB/WGP; two
//   512-thread workgroups co-resident per WGP -> 8 waves/SIMD32 to hide LDS
//   latency). All 70 gates (42 RX butterflies + 28 CNOT swaps) are applied as
//   LDS-resident sweeps with workgroup barriers. |amp|^2 written to d_ws
//   (8 MB, stays resident in the 192 MB L2).
//
// Kernel 2 (zexp_wmma_kernel): Z readout as a GEMM on the matrix core:
//   Z[128,14] = P[128,16384] x S[16384,16-pad], S[s,n] = 1 - 2*bit_{13-n}(s).
//   V_WMMA_F32_16X16X4_F32 with the +/-1 sign operand synthesized in registers
//   (branchless cndmask, no EXEC churn; pad columns 14-15 accumulate garbage
//   and are never stored). 8 waves each own a K-slice of 2048; partial 16x16
//   accumulators are summed through LDS.
// ============================================================================

#define NQ       14
#define NLAYERS  2
#define NSTATES  16384        // 2^14
#define NPAIRS   (NSTATES/2)
#define BATCH    128
#define THREADS  512          // 16 waves; 2 WGs/WGP (256 KB of 320 KB LDS)

typedef float v2f __attribute__((ext_vector_type(2)));
typedef float v8f __attribute__((ext_vector_type(8)));

// ---------------------------------------------------------------------------
// RX(theta) on qubit q. Flattened index: qubit i <-> bit position (13 - i).
// RX = [[c, -i s],[-i s, c]], c=cos(t/2), s=sin(t/2).
// new0 = c*a0 + (-i s)*a1 ; new1 = (-i s)*a0 + c*a1   (complex, c real)
// ---------------------------------------------------------------------------
__device__ __forceinline__ void apply_rx(float* __restrict__ sre,
                                         float* __restrict__ sim,
                                         float theta, int q, int tid) {
    const int bp = 13 - q;
    const float c = __builtin_cosf(0.5f * theta);
    const float s = __builtin_sinf(0.5f * theta);
    const int lowmask = (1 << bp) - 1;
    for (int p = tid; p < NPAIRS; p += THREADS) {
        const int i0 = ((p >> bp) << (bp + 1)) | (p & lowmask);
        const int i1 = i0 | (1 << bp);
        const float r0 = sre[i0], m0 = sim[i0];
        const float r1 = sre[i1], m1 = sim[i1];
        sre[i0] = c * r0 + s * m1;     // Re(c*a0 - i s*a1)
        sim[i0] = c * m0 - s * r1;     // Im(c*a0 - i s*a1)
        sre[i1] = s * m0 + c * r1;     // Re(-i s*a0 + c*a1)
        sim[i1] = c * m1 - s * r0;     // Im(-i s*a0 + c*a1)
    }
    __syncthreads();
}

// ---------------------------------------------------------------------------
// CNOT(control cq, target tq): swap amplitudes (c=1,t=0) <-> (c=1,t=1).
// Enumerate 2^12 base indices with both bits removed, then re-insert.
// ---------------------------------------------------------------------------
__device__ __forceinline__ void apply_cnot(float* __restrict__ sre,
                                           float* __restrict__ sim,
                                           int cq, int tq, int tid) {
    const int cp = 13 - cq, tp = 13 - tq;
    const int lo = cp < tp ? cp : tp;
    const int hi = cp < tp ? tp : cp;
    const int lomask = (1 << lo) - 1;
    const int himask = (1 << hi) - 1;
    for (int p = tid; p < NSTATES / 4; p += THREADS) {
        const int t1  = ((p >> lo) << (lo + 1)) | (p & lomask);   // insert 0 at lo
        const int idx = ((t1 >> hi) << (hi + 1)) | (t1 & himask); // insert 0 at hi
        const int iA = idx | (1 << cp);              // control=1, target=0
        const int iB = iA  | (1 << tp);              // control=1, target=1
        float r = sre[iA]; sre[iA] = sre[iB]; sre[iB] = r;
        float m = sim[iA]; sim[iA] = sim[iB]; sim[iB] = m;
    }
    __syncthreads();
}

// ---------------------------------------------------------------------------
// Kernel 1: full circuit, statevector in LDS, probabilities out to d_ws.
// ---------------------------------------------------------------------------
__global__ void __launch_bounds__(THREADS)
qsim_kernel(const float* __restrict__ x,       // [128,14]
            const float* __restrict__ params,  // [2,14]
            float* __restrict__ probs) {       // [128,16384] scratch
    extern __shared__ float smem[];
    float* sre  = smem;            // 16384 floats
    float* simg = smem + NSTATES;  // 16384 floats
    const int b   = blockIdx.x;
    const int tid = threadIdx.x;

    // |0...0>
    for (int i = tid; i < NSTATES; i += THREADS) { sre[i] = 0.f; simg[i] = 0.f; }
    __syncthreads();
    if (tid == 0) sre[0] = 1.f;
    __syncthreads();

    // encoding RX (per-sample angles)
    for (int i = 0; i < NQ; ++i)
        apply_rx(sre, simg, x[b * NQ + i], i, tid);

    // variational layers
    for (int layer = 0; layer < NLAYERS; ++layer) {
        for (int i = 0; i < NQ; ++i)
            apply_rx(sre, simg, params[layer * NQ + i], i, tid);
        for (int i = 0; i < NQ - 1; ++i)
            apply_cnot(sre, simg, i, i + 1, tid);
        apply_cnot(sre, simg, NQ - 1, 0, tid);
    }

    // probabilities -> global (coalesced; stays resident in 192MB L2)
    float* pout = probs + (size_t)b * NSTATES;
    for (int i = tid; i < NSTATES; i += THREADS) {
        const float r = sre[i], m = simg[i];
        pout[i] = r * r + m * m;
    }
}

// ---------------------------------------------------------------------------
// Kernel 2: Z[sample, qubit] = sum_s P[sample,s] * (1 - 2*bit_{13-qubit}(s))
// as V_WMMA_F32_16X16X4_F32 GEMM. Block = 16 samples x 16 cols (14 used).
// Wave w accumulates K-slice [w*2048, (w+1)*2048); LDS tree-sum of 8 partials.
//
// Operand layouts (ISA 7.12.2, wave32):
//   A 16x4 : lane L, vgpr v -> A[M = L%16, K = 2*(L/16) + v]
//   B 4x16 : lane L, vgpr v -> B[K = v + 2*(L/16), N = L%16]  (row-striped)
//   C/D    : lane L, vgpr r -> D[M = r + 8*(L/16), N = L%16]
// ---------------------------------------------------------------------------
__global__ void __launch_bounds__(256)
zexp_wmma_kernel(const float* __restrict__ probs,  // [128,16384]
                 float* __restrict__ out) {        // [128,14]
    __shared__ float partial[8 * 256];
    const int tid  = threadIdx.x;
    const int lane = tid & 31;
    const int wave = tid >> 5;
    const int m    = lane & 15;   // A row (sample) / B-D column (qubit)
    const int h    = lane >> 4;   // lane half
    const int sampleBase = blockIdx.x * 16;

    const float* Arow = probs + (size_t)(sampleBase + m) * NSTATES;

    // Sign bit position for qubit column m; clamped for pad columns 14/15
    // (their D elements are never stored, so garbage signs are harmless and
    // we avoid any divergent branch / EXEC manipulation in the WMMA loop).
    const int bp = (m < NQ) ? (13 - m) : 0;

    v8f acc = {};
    const int kBegin = wave * (NSTATES / 8);
    const int kEnd   = kBegin + (NSTATES / 8);

    for (int k0 = kBegin; k0 < kEnd; k0 += 4) {
        const int ka = k0 + 2 * h;      // this lane covers K = ka, ka+1 (A and B)

        // A operand: two contiguous probabilities (8B-aligned vector load)
        const v2f a = *(const v2f*)(Arow + ka);

        // B operand: branchless +/-1 signs (v_cndmask), no EXEC churn
        v2f bb;
        bb.x = ((ka >> bp) & 1)       ? -1.f : 1.f;
        bb.y = (((ka + 1) >> bp) & 1) ? -1.f : 1.f;

        // D = A(16x4) * B(4x16) + C   -- v_wmma_f32_16x16x4_f32
        acc = __builtin_amdgcn_wmma_f32_16x16x4_f32(
            /*neg_a=*/false, a, /*neg_b=*/false, bb,
            /*c_mod=*/(short)0, acc, /*reuse_a=*/false, /*reuse_b=*/false);
    }

    // spill partial tile to LDS: [wave][vgpr r][lane]
#pragma unroll
    for (int r = 0; r < 8; ++r)
        partial[wave * 256 + r * 32 + lane] = acc[r];
    __syncthreads();

    // each of the 256 threads owns one D element; sum the 8 wave partials
    float sum = 0.f;
#pragma unroll
    for (int w = 0; w < 8; ++w)
        sum += partial[w * 256 + tid];

    const int r  = tid >> 5;
    const int l2 = tid & 31;
    const int M  = r + 8 * (l2 >> 4);   // sample row within tile
    const int N  = l2 & 15;             // qubit column
    if (N < NQ)
        out[(size_t)(sampleBase + M) * NQ + N] = sum;
}

// ---------------------------------------------------------------------------
extern "C" void kernel_launch(void* const* d_in, const int* in_sizes, int n_in,
                              void* d_out, int out_size, void* d_ws, size_t ws_size,
                              hipStream_t stream) {
    (void)in_sizes; (void)n_in; (void)out_size; (void)ws_size;
    const float* x      = (const float*)d_in[0];   // [128,14]
    const float* params = (const float*)d_in[1];   // [2,14]
    float* probs = (float*)d_ws;                   // needs 128*16384*4 = 8 MB
    float* z     = (float*)d_out;                  // [128,14]

    // Kernel 1: one block per sample, 128 KB dynamic LDS statevector.
    qsim_kernel<<<BATCH, THREADS, 2 * NSTATES * sizeof(float), stream>>>(x, params, probs);

    // Kernel 2: 8 blocks x (16 samples x 16 qubit-cols), WMMA reduction.
    zexp_wmma_kernel<<<BATCH / 16, 256, 0, stream>>>(probs, z);
}